// DApp_classifier_87643102642497
// MI455X (gfx1250) — compile-verified
//
#include <hip/hip_runtime.h>
#include <hip/hip_bf16.h>

#define N_NODES   50000
#define N_EDGES   1600000
#define D         64
#define N_GRAPHS  512
#define N_CLASSES 53
#define ITERS     3
#define BN_EPS    1e-5f

typedef __attribute__((ext_vector_type(8))) float v8f;
typedef __attribute__((ext_vector_type(2))) float v2f;

// ---------------------------------------------------------------- utilities
__global__ void zero_f32(float* __restrict__ p, int n) {
    int i = blockIdx.x * blockDim.x + threadIdx.x;
    if (i < n) p[i] = 0.0f;
}

// h[i, :] = emb[pkt[i], :]
__global__ void embed_kernel(const int* __restrict__ pkt,
                             const float* __restrict__ emb,
                             float* __restrict__ h) {
    int idx = blockIdx.x * blockDim.x + threadIdx.x;
    if (idx < N_NODES * D) {
        int i = idx >> 6, c = idx & 63;
        h[idx] = emb[pkt[i] * D + c];
    }
}

// xa = (1 + eps) * h
__global__ void agg_init_kernel(const float* __restrict__ h,
                                const float* __restrict__ eps,
                                float* __restrict__ xa) {
    int idx = blockIdx.x * blockDim.x + threadIdx.x;
    if (idx < N_NODES * D) xa[idx] = (1.0f + eps[0]) * h[idx];
}

// xa[dst[e], :] += h[src[e], :]   (one wave32 per edge, 2 floats/lane)
__global__ __launch_bounds__(256) void scatter_kernel(const int* __restrict__ src,
                                                      const int* __restrict__ dst,
                                                      const float* __restrict__ h,
                                                      float* __restrict__ xa) {
    int e    = blockIdx.x * 8 + (threadIdx.x >> 5);
    int lane = threadIdx.x & 31;
    if (e < N_EDGES) {
        int s = src[e], d = dst[e];
        const float2* hp = (const float2*)(h + (size_t)s * D);
        float2 v = hp[lane];
        unsafeAtomicAdd(&xa[(size_t)d * D + lane * 2 + 0], v.x);
        unsafeAtomicAdd(&xa[(size_t)d * D + lane * 2 + 1], v.y);
    }
}

// ------------------------------------------------ fp32 WMMA MLP layer + ReLU
// y = relu(x @ W + b), x:[nrows,64], W:[64,64], b:[64]
// 256 threads = 8 waves; wave w owns rows [blk*128 + 16w, +16).
__global__ __launch_bounds__(256) void mlp_layer_kernel(const float* __restrict__ x,
                                                        const float* __restrict__ W,
                                                        const float* __restrict__ b,
                                                        float* __restrict__ y,
                                                        int nrows) {
    __shared__ float sW[D * D];      // 16 KB
    __shared__ float sB[D];
    __shared__ float sA[8][16 * D];  // 32 KB (16x64 tile per wave)

    int tid = threadIdx.x;
    for (int i = tid; i < D * D; i += 256) sW[i] = W[i];
    if (tid < D) sB[tid] = b[tid];

    int w       = tid >> 5;
    int lane    = tid & 31;
    int rowbase = blockIdx.x * 128 + w * 16;

    // stage A tile with coalesced float4 loads (zeros past nrows)
    const float4* xin = (const float4*)x;
    float4* sa4 = (float4*)sA[w];
    #pragma unroll
    for (int it = 0; it < 8; ++it) {
        int idx4 = lane + 32 * it;        // 0..255 float4s in the 16x64 tile
        int r    = idx4 >> 4;             // tile row
        int row  = rowbase + r;
        float4 v = make_float4(0.f, 0.f, 0.f, 0.f);
        if (row < nrows) v = xin[(size_t)row * 16 + (idx4 & 15)];
        sa4[idx4] = v;
    }
    __syncthreads();

    int half = lane >> 4;   // 0: K even pair, 1: K odd pair (per ISA A layout)
    int lm   = lane & 15;

    v8f acc[4];
    #pragma unroll
    for (int nt = 0; nt < 4; ++nt) acc[nt] = (v8f){0.f,0.f,0.f,0.f,0.f,0.f,0.f,0.f};

    const float* A = sA[w];
    #pragma unroll
    for (int kc = 0; kc < 16; ++kc) {        // K = 64 in chunks of 4
        int k0 = kc * 4 + half * 2;
        v2f a;
        a.x = A[lm * D + k0 + 0];            // A[M=lm][K=k0]
        a.y = A[lm * D + k0 + 1];            // A[M=lm][K=k0+1]
        #pragma unroll
        for (int nt = 0; nt < 4; ++nt) {
            v2f bb;
            bb.x = sW[(k0 + 0) * D + nt * 16 + lm];   // B[K=k0][N]
            bb.y = sW[(k0 + 1) * D + nt * 16 + lm];   // B[K=k0+1][N]
            acc[nt] = __builtin_amdgcn_wmma_f32_16x16x4_f32(
                false, a, false, bb, (short)0, acc[nt], false, false);
        }
    }

    // writeback (C/D layout: VGPR r -> rows r (lanes 0-15) and r+8 (lanes 16-31))
    #pragma unroll
    for (int nt = 0; nt < 4; ++nt) {
        #pragma unroll
        for (int r = 0; r < 8; ++r) {
            int row = rowbase + r + half * 8;
            int col = nt * 16 + lm;
            if (row < nrows) {
                float v = acc[nt][r] + sB[col];
                y[(size_t)row * D + col] = v > 0.f ? v : 0.f;
            }
        }
    }
}

// ------------------------------------------------------------ batchnorm stats
// stats[0:64] += sum_i x[i,c];  stats[64:128] += sum_i x[i,c]^2
__global__ __launch_bounds__(256) void bn_stats_kernel(const float* __restrict__ x,
                                                       float* __restrict__ stats,
                                                       int nrows) {
    __shared__ float red1[4][64];
    __shared__ float red2[4][64];
    int c  = threadIdx.x & 63;
    int rg = threadIdx.x >> 6;
    float s1 = 0.f, s2 = 0.f;
    for (int row = blockIdx.x * 4 + rg; row < nrows; row += gridDim.x * 4) {
        float v = x[(size_t)row * D + c];
        s1 += v;
        s2 += v * v;
    }
    red1[rg][c] = s1;
    red2[rg][c] = s2;
    __syncthreads();
    if (threadIdx.x < 64) {
        float a = red1[0][c] + red1[1][c] + red1[2][c] + red1[3][c];
        float q = red2[0][c] + red2[1][c] + red2[2][c] + red2[3][c];
        unsafeAtomicAdd(&stats[c],      a);
        unsafeAtomicAdd(&stats[64 + c], q);
    }
}

// normalize in place + scatter-add into pooled[:, t*64 : t*64+64]
__global__ void bn_norm_pool_kernel(float* __restrict__ h,
                                    const float* __restrict__ stats,
                                    const float* __restrict__ gamma,
                                    const float* __restrict__ beta,
                                    const int* __restrict__ gid,
                                    float* __restrict__ pooled,
                                    int t) {
    int idx = blockIdx.x * blockDim.x + threadIdx.x;
    if (idx >= N_NODES * D) return;
    int i = idx >> 6, c = idx & 63;
    float mean = stats[c] * (1.0f / N_NODES);
    float var  = stats[64 + c] * (1.0f / N_NODES) - mean * mean;
    float val  = gamma[c] * (h[idx] - mean) * rsqrtf(var + BN_EPS) + beta[c];
    h[idx] = val;
    unsafeAtomicAdd(&pooled[(size_t)gid[i] * (ITERS * D) + t * D + c], val);
}

// out = pooled @ Wc + bc   (tiny: 512x192x53)
__global__ void classifier_kernel(const float* __restrict__ pooled,
                                  const float* __restrict__ Wc,
                                  const float* __restrict__ bc,
                                  float* __restrict__ out) {
    int idx = blockIdx.x * blockDim.x + threadIdx.x;
    if (idx >= N_GRAPHS * N_CLASSES) return;
    int g = idx / N_CLASSES, k = idx % N_CLASSES;
    float acc = bc[k];
    const float* p = pooled + (size_t)g * (ITERS * D);
    for (int j = 0; j < ITERS * D; ++j) acc += p[j] * Wc[j * N_CLASSES + k];
    out[idx] = acc;
}

// ---------------------------------------------------------------------------
extern "C" void kernel_launch(void* const* d_in, const int* in_sizes, int n_in,
                              void* d_out, int out_size, void* d_ws, size_t ws_size,
                              hipStream_t stream) {
    (void)in_sizes; (void)n_in; (void)out_size; (void)ws_size;
    const int*   pkt   = (const int*)  d_in[0];
    const int*   src   = (const int*)  d_in[1];
    const int*   dst   = (const int*)  d_in[2];
    const int*   gid   = (const int*)  d_in[3];
    const float* emb   = (const float*)d_in[4];
    const float* eps   = (const float*)d_in[5];
    const float* W1    = (const float*)d_in[6];
    const float* b1    = (const float*)d_in[7];
    const float* W2    = (const float*)d_in[8];
    const float* b2    = (const float*)d_in[9];
    const float* W3    = (const float*)d_in[10];
    const float* b3    = (const float*)d_in[11];
    const float* gamma = (const float*)d_in[12];
    const float* beta  = (const float*)d_in[13];
    const float* Wc    = (const float*)d_in[14];
    const float* bc    = (const float*)d_in[15];
    float* out = (float*)d_out;

    char* ws = (char*)d_ws;
    float* h      = (float*)(ws);                 // 50000*64*4 = 12,800,000 B
    float* xa     = (float*)(ws + 12800000);      // 12,800,000 B
    float* pooled = (float*)(ws + 25600000);      // 512*192*4 = 393,216 B
    float* stats  = (float*)(ws + 25993216);      // 128*4 B

    const int ND = N_NODES * D;
    zero_f32<<<(N_GRAPHS * ITERS * D + 255) / 256, 256, 0, stream>>>(pooled, N_GRAPHS * ITERS * D);
    embed_kernel<<<(ND + 255) / 256, 256, 0, stream>>>(pkt, emb, h);

    int mlpGrid = (N_NODES + 127) / 128;
    for (int t = 0; t < ITERS; ++t) {
        agg_init_kernel<<<(ND + 255) / 256, 256, 0, stream>>>(h, eps, xa);
        scatter_kernel<<<(N_EDGES + 7) / 8, 256, 0, stream>>>(src, dst, h, xa);
        mlp_layer_kernel<<<mlpGrid, 256, 0, stream>>>(xa, W1, b1, h,  N_NODES);
        mlp_layer_kernel<<<mlpGrid, 256, 0, stream>>>(h,  W2, b2, xa, N_NODES);
        mlp_layer_kernel<<<mlpGrid, 256, 0, stream>>>(xa, W3, b3, h,  N_NODES);
        zero_f32<<<1, 128, 0, stream>>>(stats, 128);
        bn_stats_kernel<<<256, 256, 0, stream>>>(h, stats, N_NODES);
        bn_norm_pool_kernel<<<(ND + 255) / 256, 256, 0, stream>>>(h, stats, gamma, beta, gid, pooled, t);
    }
    classifier_kernel<<<(N_GRAPHS * N_CLASSES + 255) / 256, 256, 0, stream>>>(pooled, Wc, bc, out);
}